// GraphSAINTNet_317827580335
// MI455X (gfx1250) — compile-verified
//
#include <hip/hip_runtime.h>

// ---------------------------------------------------------------------------
// GraphSAINT 2-layer GCN for MI455X (gfx1250, wave32).
//   h1 = x @ W1                         (WMMA f32 16x16x4)
//   a1 = GCN-aggregate(h1) + b1        (CSR gather, no float atomics)
//   a1 = relu(batchnorm(a1))           (column stats + in-place apply)
//   g2 = a1 @ W2                        (WMMA f32 16x16x4)
//   out = GCN-aggregate(g2) + b2
// ---------------------------------------------------------------------------

typedef float v2f __attribute__((ext_vector_type(2)));
typedef float v8f __attribute__((ext_vector_type(8)));

#define IN_F  128
#define HID_F 256
#define OUT_F 64
#define BN_EPS 1e-5f

// ---------------------------- small utility kernels ------------------------

__global__ void zero_kernel(int* __restrict__ deg, float* __restrict__ stats, int N) {
    int i = blockIdx.x * blockDim.x + threadIdx.x;
    if (i < N)   deg[i] = 0;
    if (i < 2 * HID_F) stats[i] = 0.0f;
}

__global__ void deg_kernel(const int* __restrict__ dst, int* __restrict__ deg, int E) {
    int e = blockIdx.x * blockDim.x + threadIdx.x;
    if (e < E) atomicAdd(&deg[dst[e]], 1);
}

// dis[i] = rsqrt(in_degree + 1 self loop)
__global__ void dis_kernel(const int* __restrict__ deg, float* __restrict__ dis, int N) {
    int i = blockIdx.x * blockDim.x + threadIdx.x;
    if (i < N) dis[i] = rsqrtf((float)deg[i] + 1.0f);
}

// Per-block exclusive scan (Hillis-Steele on 256 elements), block totals out.
__global__ void scan_block_kernel(const int* __restrict__ deg, int* __restrict__ rs,
                                  int* __restrict__ bsum, int N) {
    __shared__ int sh[256];
    int i = blockIdx.x * 256 + threadIdx.x;
    int v = (i < N) ? deg[i] : 0;
    sh[threadIdx.x] = v;
    __syncthreads();
    for (int off = 1; off < 256; off <<= 1) {
        int t = (threadIdx.x >= off) ? sh[threadIdx.x - off] : 0;
        __syncthreads();
        sh[threadIdx.x] += t;
        __syncthreads();
    }
    if (i < N) rs[i] = sh[threadIdx.x] - v;          // exclusive within block
    if (threadIdx.x == 255) bsum[blockIdx.x] = sh[255];
}

// Serial scan of ~N/256 block sums (tiny) + write total to rs[N].
__global__ void scan_bsum_kernel(int* __restrict__ bsum, int* __restrict__ rs,
                                 int nblk, int N) {
    if (blockIdx.x == 0 && threadIdx.x == 0) {
        int run = 0;
        for (int b = 0; b < nblk; ++b) { int t = bsum[b]; bsum[b] = run; run += t; }
        rs[N] = run;
    }
}

__global__ void scan_add_kernel(int* __restrict__ rs, const int* __restrict__ bsum,
                                int* __restrict__ pos, int N) {
    int i = blockIdx.x * blockDim.x + threadIdx.x;
    if (i < N) {
        int v = rs[i] + bsum[i >> 8];
        rs[i]  = v;
        pos[i] = v;
    }
}

__global__ void scatter_kernel(const int* __restrict__ src, const int* __restrict__ dst,
                               int* __restrict__ pos, int* __restrict__ esrc, int E) {
    int e = blockIdx.x * blockDim.x + threadIdx.x;
    if (e < E) {
        int p = atomicAdd(&pos[dst[e]], 1);
        esrc[p] = src[e];
    }
}

// ------------------------------ WMMA GEMM ----------------------------------
// C[N x NOUT] = A[N x K] * B[K x NOUT], fp32 via V_WMMA_F32_16X16X4_F32.
// Block: 128 threads (4 waves). Block computes 16 rows x NOUT cols.
// Each wave owns NOUT/16/4 column tiles; K streamed through LDS in chunks.
template <int K, int NOUT>
__global__ void gemm_wmma_kernel(const float* __restrict__ A, const float* __restrict__ B,
                                 float* __restrict__ C, int N) {
    constexpr int KC  = 32;
    constexpr int TPW = (NOUT / 16) / 4;     // col tiles per wave
    __shared__ float sA[16 * KC];            // 2 KB
    __shared__ float sB[KC * NOUT];          // <= 32 KB

    const int row0 = blockIdx.x * 16;
    const int wave = threadIdx.x >> 5;
    const int lane = threadIdx.x & 31;
    const int m    = lane & 15;              // matrix row within fragment
    const int kh   = (lane >> 4) * 2;        // K half select (0 or 2) per ISA layout

    v8f acc[TPW];
    for (int t = 0; t < TPW; ++t) acc[t] = (v8f){0,0,0,0,0,0,0,0};

    for (int k0 = 0; k0 < K; k0 += KC) {
        // Stage A tile (zero-fill OOB rows so EXEC stays full for WMMA).
        for (int i = threadIdx.x; i < 16 * KC; i += 128) {
            int r = i / KC, c = i % KC;
            int gr = row0 + r;
            sA[i] = (gr < N) ? A[(size_t)gr * K + k0 + c] : 0.0f;
        }
        // Stage B tile.
        for (int i = threadIdx.x; i < KC * NOUT; i += 128) {
            int r = i / NOUT, c = i % NOUT;
            sB[i] = B[(size_t)(k0 + r) * NOUT + c];
        }
        __syncthreads();

        for (int kk = 0; kk < KC; kk += 4) {
            // A frag: lanes 0-15 hold K=kk,kk+1; lanes 16-31 hold K=kk+2,kk+3.
            v2f af;
            af.x = sA[m * KC + kk + kh];
            af.y = sA[m * KC + kk + kh + 1];
            for (int t = 0; t < TPW; ++t) {
                int n0 = (wave * TPW + t) * 16;
                v2f bf;                       // B frag mirrors A's K striping across lanes
                bf.x = sB[(kk + kh)     * NOUT + n0 + m];
                bf.y = sB[(kk + kh + 1) * NOUT + n0 + m];
                acc[t] = __builtin_amdgcn_wmma_f32_16x16x4_f32(
                    /*neg_a=*/false, af, /*neg_b=*/false, bf,
                    /*c_mod=*/(short)0, acc[t], /*reuse_a=*/false, /*reuse_b=*/false);
            }
        }
        __syncthreads();
    }

    // C/D layout: VGPR j -> M = j (lanes 0-15) / M = 8+j (lanes 16-31), N = lane&15.
    const int rbase = (lane >> 4) * 8;
    for (int t = 0; t < TPW; ++t) {
        int col = (wave * TPW + t) * 16 + m;
        for (int j = 0; j < 8; ++j) {
            int r = row0 + rbase + j;
            if (r < N) C[(size_t)r * NOUT + col] = acc[t][j];
        }
    }
}

// ------------------------ CSR gather aggregation ---------------------------
// One wave per node. out[i] = dis[i] * (sum_{s in N(i)} dis[s]*H[s] + dis[i]*H[i]) + bias.
template <int C>
__global__ void agg_kernel(const float* __restrict__ H, const float* __restrict__ dis,
                           const int* __restrict__ rs, const int* __restrict__ esrc,
                           const float* __restrict__ bias, float* __restrict__ out, int N) {
    int wid = (blockIdx.x * blockDim.x + threadIdx.x) >> 5;
    if (wid >= N) return;
    const int lane = threadIdx.x & 31;
    constexpr int V = C / 32;                 // channels per lane (8 or 2)
    const int cbase = lane * V;

    const float di = dis[wid];
    float acc[V];
    const float* hr = H + (size_t)wid * C + cbase;
    #pragma unroll
    for (int j = 0; j < V; ++j) acc[j] = di * hr[j];   // self loop term

    const int p0 = rs[wid], p1 = rs[wid + 1];
    for (int p = p0; p < p1; ++p) {
        int s = esrc[p];
        if (p + 1 < p1) {                      // prefetch next gather row (global_prefetch_b8)
            int sn = esrc[p + 1];
            __builtin_prefetch(H + (size_t)sn * C + cbase, 0, 1);
        }
        float w = dis[s];
        const float* hs = H + (size_t)s * C + cbase;
        #pragma unroll
        for (int j = 0; j < V; ++j) acc[j] += w * hs[j];
    }

    float* o = out + (size_t)wid * C + cbase;
    #pragma unroll
    for (int j = 0; j < V; ++j) o[j] = di * acc[j] + bias[cbase + j];
}

// ------------------------------ BatchNorm ----------------------------------

__global__ void bn_stats_kernel(const float* __restrict__ a1, float* __restrict__ stats, int N) {
    const int col = threadIdx.x;              // 256 threads = 256 columns
    int r0 = blockIdx.x * 512;
    int r1 = r0 + 512; if (r1 > N) r1 = N;
    float s = 0.0f, s2 = 0.0f;
    for (int r = r0; r < r1; ++r) {
        float v = a1[(size_t)r * HID_F + col];
        s += v; s2 += v * v;
    }
    atomicAdd(&stats[col], s);
    atomicAdd(&stats[HID_F + col], s2);
}

__global__ void bn_apply_kernel(float* __restrict__ a1, const float* __restrict__ stats,
                                const float* __restrict__ gamma, const float* __restrict__ beta,
                                int N) {
    size_t i = (size_t)blockIdx.x * blockDim.x + threadIdx.x;
    if (i >= (size_t)N * HID_F) return;
    int col = (int)(i & (HID_F - 1));
    float inv_n = 1.0f / (float)N;
    float mean  = stats[col] * inv_n;
    float var   = stats[HID_F + col] * inv_n - mean * mean;
    float v = (a1[i] - mean) * rsqrtf(var + BN_EPS) * gamma[col] + beta[col];
    a1[i] = v > 0.0f ? v : 0.0f;
}

// ------------------------------- launcher ----------------------------------

extern "C" void kernel_launch(void* const* d_in, const int* in_sizes, int n_in,
                              void* d_out, int out_size, void* d_ws, size_t ws_size,
                              hipStream_t stream) {
    const float* x     = (const float*)d_in[0];
    const int*   edge  = (const int*)  d_in[1];   // [2, E] row-major
    const float* W1    = (const float*)d_in[2];
    const float* b1    = (const float*)d_in[3];
    const float* gamma = (const float*)d_in[4];
    const float* beta  = (const float*)d_in[5];
    const float* W2    = (const float*)d_in[6];
    const float* b2    = (const float*)d_in[7];
    float* out = (float*)d_out;

    const int N = in_sizes[0] / IN_F;
    const int E = in_sizes[1] / 2;
    const int* src = edge;
    const int* dst = edge + E;
    const int nblk = (N + 255) / 256;

    // Workspace carve-up (all 4-byte types; base is 256B aligned).
    char* p = (char*)d_ws;
    float* h1    = (float*)p; p += (size_t)N * HID_F * 4;   // x@W1, later reused as g2
    float* a1    = (float*)p; p += (size_t)N * HID_F * 4;   // aggregated layer-1 output
    float* dis   = (float*)p; p += (size_t)N * 4;
    float* stats = (float*)p; p += 2 * HID_F * 4;
    int*   deg   = (int*)p;   p += (size_t)N * 4;
    int*   rs    = (int*)p;   p += ((size_t)N + 1) * 4;
    int*   bsum  = (int*)p;   p += (size_t)nblk * 4;
    int*   pos   = (int*)p;   p += (size_t)N * 4;
    int*   esrc  = (int*)p;   p += (size_t)E * 4;

    const int TB = 256;
    const int gN = (N + TB - 1) / TB;
    const int gE = (E + TB - 1) / TB;

    // 1) degrees + norm
    zero_kernel<<<gN, TB, 0, stream>>>(deg, stats, N);
    deg_kernel<<<gE, TB, 0, stream>>>(dst, deg, E);
    dis_kernel<<<gN, TB, 0, stream>>>(deg, dis, N);

    // 2) CSR by destination
    scan_block_kernel<<<nblk, 256, 0, stream>>>(deg, rs, bsum, N);
    scan_bsum_kernel<<<1, 32, 0, stream>>>(bsum, rs, nblk, N);
    scan_add_kernel<<<gN, TB, 0, stream>>>(rs, bsum, pos, N);
    scatter_kernel<<<gE, TB, 0, stream>>>(src, dst, pos, esrc, E);

    // 3) layer 1: GEMM (WMMA), aggregate, BN + ReLU
    int gemm_blocks = (N + 15) / 16;
    gemm_wmma_kernel<IN_F, HID_F><<<gemm_blocks, 128, 0, stream>>>(x, W1, h1, N);
    int agg_blocks = (N + 7) / 8;                         // 8 waves/block, 1 wave/node
    agg_kernel<HID_F><<<agg_blocks, 256, 0, stream>>>(h1, dis, rs, esrc, b1, a1, N);
    bn_stats_kernel<<<(N + 511) / 512, 256, 0, stream>>>(a1, stats, N);
    size_t tot1 = (size_t)N * HID_F;
    bn_apply_kernel<<<(unsigned)((tot1 + TB - 1) / TB), TB, 0, stream>>>(a1, stats, gamma, beta, N);

    // 4) layer 2: GEMM (WMMA) into reused h1 buffer, aggregate into d_out
    float* g2 = h1;
    gemm_wmma_kernel<HID_F, OUT_F><<<gemm_blocks, 128, 0, stream>>>(a1, W2, g2, N);
    agg_kernel<OUT_F><<<agg_blocks, 256, 0, stream>>>(g2, dis, rs, esrc, b2, out, N);

    (void)n_in; (void)out_size; (void)ws_size;
}